// HeteroGATLayer_14259291423309
// MI455X (gfx1250) — compile-verified
//
#include <hip/hip_runtime.h>
#include <hip/hip_bf16.h>

#define D 64

typedef float v2f __attribute__((ext_vector_type(2)));
typedef float v8f __attribute__((ext_vector_type(8)));

// ---------------------------------------------------------------------------
// Dense projection: Wh = feat @ W^T + b  via V_WMMA_F32_16X16X4_F32.
// One wave computes a 16(row) x 64(col) output tile: 4 accumulators, K=64 in
// 16 steps of k=4.  A layout (16x4 f32): lanes 0-15 = rows, VGPR0/1 = k0/k1;
// lanes 16-31 = rows, k2/k3.  B layout mirrored (lane = col, k in VGPRs).
// B[k][j] = W[j*64 + k] (W is row-major DxD, we need W^T).
// ---------------------------------------------------------------------------
__global__ __launch_bounds__(32) void proj_wmma(const float* __restrict__ feat,
                                                const float* __restrict__ W,
                                                const float* __restrict__ b,
                                                float* __restrict__ out, int nrows) {
    const int tile = blockIdx.x;
    const int lane = threadIdx.x;
    const int rl   = lane & 15;
    const int hi   = (lane >> 4) & 1;          // 0: k0/k1 half, 1: k2/k3 half
    const int row0 = tile * 16;
    if (row0 >= nrows) return;

    const float* arow = feat + (size_t)(row0 + rl) * D + 2 * hi;
    const float* w0 = W + (size_t)(0 * 16 + rl) * D + 2 * hi;
    const float* w1 = W + (size_t)(1 * 16 + rl) * D + 2 * hi;
    const float* w2 = W + (size_t)(2 * 16 + rl) * D + 2 * hi;
    const float* w3 = W + (size_t)(3 * 16 + rl) * D + 2 * hi;

    v8f acc0 = {}, acc1 = {}, acc2 = {}, acc3 = {};
#pragma unroll
    for (int kk = 0; kk < 16; ++kk) {
        v2f a  = *(const v2f*)(arow + 4 * kk);
        v2f b0 = *(const v2f*)(w0 + 4 * kk);
        v2f b1 = *(const v2f*)(w1 + 4 * kk);
        v2f b2 = *(const v2f*)(w2 + 4 * kk);
        v2f b3 = *(const v2f*)(w3 + 4 * kk);
        acc0 = __builtin_amdgcn_wmma_f32_16x16x4_f32(false, a, false, b0, (short)0, acc0, false, false);
        acc1 = __builtin_amdgcn_wmma_f32_16x16x4_f32(false, a, false, b1, (short)0, acc1, false, false);
        acc2 = __builtin_amdgcn_wmma_f32_16x16x4_f32(false, a, false, b2, (short)0, acc2, false, false);
        acc3 = __builtin_amdgcn_wmma_f32_16x16x4_f32(false, a, false, b3, (short)0, acc3, false, false);
    }

    // C/D layout: VGPR r, lanes 0-15 -> (M=r, N=lane); lanes 16-31 -> (M=r+8).
#pragma unroll
    for (int c = 0; c < 4; ++c) {
        v8f acc = (c == 0) ? acc0 : (c == 1) ? acc1 : (c == 2) ? acc2 : acc3;
        float bias = b[c * 16 + rl];
#pragma unroll
        for (int r = 0; r < 8; ++r) {
            int row = row0 + r + 8 * hi;
            out[(size_t)row * D + c * 16 + rl] = acc[r] + bias;
        }
    }
}

// ---------------------------------------------------------------------------
// Utility fills
// ---------------------------------------------------------------------------
__global__ void fill_f32(float* __restrict__ p, float v, size_t n) {
    size_t i = (size_t)blockIdx.x * blockDim.x + threadIdx.x;
    if (i < n) p[i] = v;
}
__global__ void fill_u32(unsigned* __restrict__ p, unsigned v, size_t n) {
    size_t i = (size_t)blockIdx.x * blockDim.x + threadIdx.x;
    if (i < n) p[i] = v;
}

// ---------------------------------------------------------------------------
// Per-node attention scalar: s[n] = dot(Wh[n,:], avec)
// ---------------------------------------------------------------------------
__global__ void node_dot(const float* __restrict__ Wh, const float* __restrict__ avec,
                         float* __restrict__ s, int n) {
    int i = blockIdx.x * blockDim.x + threadIdx.x;
    if (i >= n) return;
    const float4* r = (const float4*)(Wh + (size_t)i * D);
    const float4* a = (const float4*)avec;
    float acc = 0.f;
#pragma unroll
    for (int k = 0; k < D / 4; ++k) {
        float4 x = r[k], y = a[k];
        acc += x.x * y.x + x.y * y.y + x.z * y.z + x.w * y.w;
    }
    s[i] = acc;
}

// c1 = dot(fcW, a_e), c2 = dot(fcb, a_e): lets zij contribution be w*c1+c2
__global__ void edge_const(const float* __restrict__ fcW, const float* __restrict__ fcb,
                           const float* __restrict__ ae, float* __restrict__ c12) {
    if (threadIdx.x == 0 && blockIdx.x == 0) {
        float a = 0.f, bb = 0.f;
        for (int j = 0; j < D; ++j) { a += fcW[j] * ae[j]; bb += fcb[j] * ae[j]; }
        c12[0] = a; c12[1] = bb;
    }
}

// monotone float<->uint mapping for atomicMax-based segment max
__device__ __forceinline__ unsigned enc_f(float f) {
    unsigned u = __float_as_uint(f);
    return (u & 0x80000000u) ? ~u : (u | 0x80000000u);
}
__device__ __forceinline__ float dec_f(unsigned u) {
    unsigned v = (u & 0x80000000u) ? (u & 0x7FFFFFFFu) : ~u;
    return __uint_as_float(v);
}

// pass 1: logits + segment max
__global__ void edge_logits(const int* __restrict__ src, const int* __restrict__ dst,
                            const float* __restrict__ s1, const float* __restrict__ s2,
                            const float* __restrict__ wf, const float* __restrict__ c12,
                            float* __restrict__ e, unsigned* __restrict__ menc, int E) {
    int i = blockIdx.x * blockDim.x + threadIdx.x;
    if (i >= E) return;
    float v = s1[src[i]] + s2[dst[i]];
    if (wf) v += wf[i] * c12[0] + c12[1];
    v = v > 0.f ? v : 0.2f * v;       // leaky_relu(., 0.2)
    e[i] = v;
    atomicMax(&menc[dst[i]], enc_f(v));
}

// pass 2: exp(e - max) + segment sum of denominators
__global__ void edge_exp(const int* __restrict__ dst, float* __restrict__ e,
                         const unsigned* __restrict__ menc, float* __restrict__ den, int E) {
    int i = blockIdx.x * blockDim.x + threadIdx.x;
    if (i >= E) return;
    int d = dst[i];
    float ex = expf(e[i] - dec_f(menc[d]));
    e[i] = ex;
    atomicAdd(&den[d], ex);
}

// pass 3: alpha-weighted scatter of (Wh_src [+ zij]); 64 threads per edge
__global__ void edge_scatter(const int* __restrict__ src, const int* __restrict__ dst,
                             const float* __restrict__ e, const float* __restrict__ den,
                             const float* __restrict__ Whs, const float* __restrict__ wf,
                             const float* __restrict__ fcW, const float* __restrict__ fcb,
                             float* __restrict__ hout, int E) {
    size_t t = (size_t)blockIdx.x * blockDim.x + threadIdx.x;
    int i = (int)(t >> 6);
    int j = (int)(t & 63);
    if (i >= E) return;
    int s = src[i], d = dst[i];
    float coef = e[i] / den[d];
    float msg = Whs[(size_t)s * D + j];
    if (wf) msg += wf[i] * fcW[j] + fcb[j];
    atomicAdd(&hout[(size_t)d * D + j], coef * msg);
}

// plain copy_src + sum (sin etype)
__global__ void sum_scatter(const int* __restrict__ src, const int* __restrict__ dst,
                            const float* __restrict__ Whs, float* __restrict__ hout, int E) {
    size_t t = (size_t)blockIdx.x * blockDim.x + threadIdx.x;
    int i = (int)(t >> 6);
    int j = (int)(t & 63);
    if (i >= E) return;
    atomicAdd(&hout[(size_t)dst[i] * D + j], Whs[(size_t)src[i] * D + j]);
}

__global__ void relu_ip(float* __restrict__ p, size_t n) {
    size_t i = (size_t)blockIdx.x * blockDim.x + threadIdx.x;
    if (i < n) p[i] = fmaxf(p[i], 0.f);
}

// ---------------------------------------------------------------------------
// Host orchestration
// ---------------------------------------------------------------------------
extern "C" void kernel_launch(void* const* d_in, const int* in_sizes, int n_in,
                              void* d_out, int out_size, void* d_ws, size_t ws_size,
                              hipStream_t stream) {
    const int N = in_sizes[0] / D;
    const int E = in_sizes[3];

    const float* feat_task   = (const float*)d_in[0];
    const float* feat_worker = (const float*)d_in[1];
    const float* feat_state  = (const float*)d_in[2];
    const int *SRC[8], *DST[8];
    for (int t = 0; t < 8; ++t) { SRC[t] = (const int*)d_in[3 + 2 * t]; DST[t] = (const int*)d_in[4 + 2 * t]; }
    const float* w_temporal = (const float*)d_in[19];
    const float* t_take     = (const float*)d_in[20];
    const float* t_use      = (const float*)d_in[21];
    const float *Wet[8], *bet[8];
    for (int t = 0; t < 8; ++t) { Wet[t] = (const float*)d_in[22 + 2 * t]; bet[t] = (const float*)d_in[23 + 2 * t]; }
    const float *fcW[3], *fcb[3];
    for (int t = 0; t < 3; ++t) { fcW[t] = (const float*)d_in[38 + 2 * t]; fcb[t] = (const float*)d_in[39 + 2 * t]; }
    const float* attn_W        = (const float*)d_in[44];
    const float* attn_ttr_W    = (const float*)d_in[45];
    const float* attn_rut_W    = (const float*)d_in[46];
    const float* attn_assigned = (const float*)d_in[47];
    const float* attn_com      = (const float*)d_in[48];
    const float* attn_tin      = (const float*)d_in[49];
    const float* attn_rin      = (const float*)d_in[50];

    // workspace carve-up (floats); needs ~ (8*N*64 + 4*N + E + 2)*4 bytes
    float* ws = (float*)d_ws;
    size_t off = 0;
    float* Wh[8];
    for (int t = 0; t < 8; ++t) { Wh[t] = ws + off; off += (size_t)N * D; }
    float* s1 = ws + off; off += N;
    float* s2 = ws + off; off += N;
    unsigned* menc = (unsigned*)(ws + off); off += N;
    float* den = ws + off; off += N;
    float* ebuf = ws + off; off += E;
    float* c12 = ws + off; off += 2;

    float* out      = (float*)d_out;
    float* h_task   = out;
    float* h_worker = out + (size_t)N * D;
    float* h_state  = out + 2 * (size_t)N * D;

    const size_t outN = 3 * (size_t)N * D;
    fill_f32<<<dim3((unsigned)((outN + 255) / 256)), 256, 0, stream>>>(out, 0.f, outN);

    // Equation (1): per-etype projections via WMMA
    // ETYPES order: temporal, assigned_to, com, tin, win, sin, take_time, use_time
    const float* featFor[8] = {feat_task, feat_task, feat_worker, feat_task,
                               feat_worker, feat_state, feat_task, feat_worker};
    for (int t = 0; t < 8; ++t)
        proj_wmma<<<dim3((unsigned)((N + 15) / 16)), 32, 0, stream>>>(featFor[t], Wet[t], bet[t], Wh[t], N);

    const unsigned nb  = (unsigned)((N + 255) / 256);
    const unsigned eb  = (unsigned)((E + 255) / 256);
    const unsigned e64 = (unsigned)(((size_t)E * 64 + 255) / 256);

    auto run_gat = [&](const float* Whs, const float* Whd, const int* src, const int* dst,
                       const float* aW, const float* wf, const float* fw, const float* fb,
                       float* hout) {
        fill_u32<<<nb, 256, 0, stream>>>(menc, 0u, (size_t)N);
        fill_f32<<<nb, 256, 0, stream>>>(den, 0.f, (size_t)N);
        node_dot<<<nb, 256, 0, stream>>>(Whs, aW, s1, N);
        node_dot<<<nb, 256, 0, stream>>>(Whd, aW + D, s2, N);
        if (wf) edge_const<<<1, 64, 0, stream>>>(fw, fb, aW + 2 * D, c12);
        edge_logits<<<eb, 256, 0, stream>>>(src, dst, s1, s2, wf, c12, ebuf, menc, E);
        edge_exp<<<eb, 256, 0, stream>>>(dst, ebuf, menc, den, E);
        edge_scatter<<<e64, 256, 0, stream>>>(src, dst, ebuf, den, Whs, wf, fw, fb, hout, E);
    };

    // temporal: task->task, edge feat w_temporal
    run_gat(Wh[0], Wh[0], SRC[0], DST[0], attn_W,        w_temporal, fcW[0], fcb[0], h_task);
    // assigned_to: task->worker (dst feats are Wh_com)
    run_gat(Wh[1], Wh[2], SRC[1], DST[1], attn_assigned, nullptr, nullptr, nullptr, h_worker);
    // com: worker->worker
    run_gat(Wh[2], Wh[2], SRC[2], DST[2], attn_com,      nullptr, nullptr, nullptr, h_worker);
    // tin: task->state (dst feats are Wh_sin)
    run_gat(Wh[3], Wh[5], SRC[3], DST[3], attn_tin,      nullptr, nullptr, nullptr, h_state);
    // win: worker->state
    run_gat(Wh[4], Wh[5], SRC[4], DST[4], attn_rin,      nullptr, nullptr, nullptr, h_state);
    // take_time: task->worker, edge feat t_take_time
    run_gat(Wh[6], Wh[2], SRC[6], DST[6], attn_ttr_W,    t_take, fcW[1], fcb[1], h_worker);
    // use_time: worker->task, edge feat t_use_time
    run_gat(Wh[7], Wh[0], SRC[7], DST[7], attn_rut_W,    t_use,  fcW[2], fcb[2], h_task);
    // sin: state->state plain sum
    sum_scatter<<<e64, 256, 0, stream>>>(SRC[5], DST[5], Wh[5], h_state, E);

    relu_ip<<<dim3((unsigned)((outN + 255) / 256)), 256, 0, stream>>>(out, outN);
}